// TokenImportanceRouter_36172214567085
// MI455X (gfx1250) — compile-verified
//
#include <hip/hip_runtime.h>

#define B_   4
#define T_   4096
#define D_   2048
#define DR_  512
#define NCH_ 64          // token chunks for pass-1 reduction
#define TCH_ (T_ / NCH_) // 64 tokens per chunk

typedef __attribute__((ext_vector_type(2))) float v2f;
typedef __attribute__((ext_vector_type(8))) float v8f;

// ---------------------------------------------------------------------------
// Pass 1a: partial column sums of hidden over token chunks.
// grid = B_ * NCH_ * (D_/4/256) = 4*64*2 = 512 blocks, 256 threads.
// Coalesced float4 loads; 134 MB streamed once from HBM.
// ---------------------------------------------------------------------------
__global__ void colsum_partial(const float4* __restrict__ hidden,
                               float4* __restrict__ partial) {
  const int dblk = blockIdx.x & 1;          // 0..1
  const int ch   = (blockIdx.x >> 1) & 63;  // 0..63
  const int b    = blockIdx.x >> 7;         // 0..3
  const int d4   = dblk * 256 + threadIdx.x; // float4 column, 0..511
  const float4* p = hidden + ((size_t)(b * T_ + ch * TCH_) * (D_ / 4)) + d4;
  float4 s = make_float4(0.f, 0.f, 0.f, 0.f);
#pragma unroll 4
  for (int t = 0; t < TCH_; ++t) {
    float4 x = p[(size_t)t * (D_ / 4)];
    s.x += x.x; s.y += x.y; s.z += x.z; s.w += x.w;
  }
  partial[((size_t)(b * NCH_ + ch) * (D_ / 4)) + d4] = s;
}

// ---------------------------------------------------------------------------
// Pass 1b: reduce chunk partials -> hbar = mean_t hidden  (B_ x D_).
// grid = B_ * (D_/4/256) = 8 blocks, 256 threads.
// ---------------------------------------------------------------------------
__global__ void colsum_final(const float4* __restrict__ partial,
                             float4* __restrict__ hbar) {
  const int dblk = blockIdx.x & 1;
  const int b    = blockIdx.x >> 1;
  const int d4   = dblk * 256 + threadIdx.x;
  float4 s = make_float4(0.f, 0.f, 0.f, 0.f);
  for (int c = 0; c < NCH_; ++c) {
    float4 x = partial[((size_t)(b * NCH_ + c) * (D_ / 4)) + d4];
    s.x += x.x; s.y += x.y; s.z += x.z; s.w += x.w;
  }
  const float inv = 1.0f / (float)T_;
  hbar[(size_t)b * (D_ / 4) + d4] =
      make_float4(s.x * inv, s.y * inv, s.z * inv, s.w * inv);
}

// ---------------------------------------------------------------------------
// kbar = hbar @ Wk^T + bk   (B_ x DR_), M padded 4->16, WMMA f32 16x16x4.
// grid = DR_/16 = 32 blocks of 32 threads (one wave each, EXEC all ones).
// A frag (16x4 f32): lane -> M = lane&15; VGPR0/1 x half -> K = 2*(lane>>4)+{0,1}
// B frag (4x16 f32): lane -> N = lane&15; same K striping.
// D (16x16 f32): VGPR i = row M=i for lanes 0..15.
// ---------------------------------------------------------------------------
__global__ void kbar_wmma(const float* __restrict__ hbar,
                          const float* __restrict__ Wk,
                          const float* __restrict__ bk,
                          float* __restrict__ kbar) {
  const int lane = threadIdx.x;
  const int m    = lane & 15;
  const int h2   = (lane >> 4) * 2;       // K offset within step: 0 or 2
  const int n0   = blockIdx.x * 16;
  const int mm   = (m < B_) ? m : 0;      // clamp padded rows (no OOB, no branch)
  const float* arow = hbar + (size_t)mm * D_;
  const float* brow = Wk + (size_t)(n0 + m) * D_;   // Wk row-major (DR_ x D_)
  v8f acc = {};
  for (int d0 = 0; d0 < D_; d0 += 4) {
    v2f a, bf;
    a.x  = arow[d0 + h2];
    a.y  = arow[d0 + h2 + 1];
    bf.x = brow[d0 + h2];
    bf.y = brow[d0 + h2 + 1];
    acc = __builtin_amdgcn_wmma_f32_16x16x4_f32(false, a, false, bf,
                                                (short)0, acc, false, false);
  }
  if (lane < 16) {
    const float bias = bk[n0 + lane];
    kbar[0 * DR_ + n0 + lane] = acc[0] + bias;
    kbar[1 * DR_ + n0 + lane] = acc[1] + bias;
    kbar[2 * DR_ + n0 + lane] = acc[2] + bias;
    kbar[3 * DR_ + n0 + lane] = acc[3] + bias;
  }
}

// ---------------------------------------------------------------------------
// v = kbar @ Wq   (B_ x D_), WMMA f32 16x16x4; last block computes c = bq.kbar.
// grid = D_/16 + 1 = 129 blocks of 32 threads.
// ---------------------------------------------------------------------------
__global__ void vproj_wmma(const float* __restrict__ kbar,
                           const float* __restrict__ Wq,
                           const float* __restrict__ bq,
                           float* __restrict__ vvec,
                           float* __restrict__ cvec) {
  const int lane = threadIdx.x;
  if (blockIdx.x == D_ / 16) {
    for (int b = 0; b < B_; ++b) {
      float s = 0.f;
      for (int r = lane; r < DR_; r += 32) s += bq[r] * kbar[b * DR_ + r];
#pragma unroll
      for (int off = 16; off > 0; off >>= 1) s += __shfl_xor(s, off, 32);
      if (lane == 0) cvec[b] = s;
    }
    return;
  }
  const int m  = lane & 15;
  const int h2 = (lane >> 4) * 2;
  const int d0 = blockIdx.x * 16;
  const int mm = (m < B_) ? m : 0;
  const float* arow = kbar + (size_t)mm * DR_;
  v8f acc = {};
  for (int r0 = 0; r0 < DR_; r0 += 4) {
    v2f a, bf;
    a.x  = arow[r0 + h2];
    a.y  = arow[r0 + h2 + 1];
    bf.x = Wq[(size_t)(r0 + h2) * D_ + d0 + m];      // B[k][n] = Wq[r][d0+n]
    bf.y = Wq[(size_t)(r0 + h2 + 1) * D_ + d0 + m];
    acc = __builtin_amdgcn_wmma_f32_16x16x4_f32(false, a, false, bf,
                                                (short)0, acc, false, false);
  }
  if (lane < 16) {
    vvec[0 * D_ + d0 + lane] = acc[0];
    vvec[1 * D_ + d0 + lane] = acc[1];
    vvec[2 * D_ + d0 + lane] = acc[2];
    vvec[3 * D_ + d0 + lane] = acc[3];
  }
}

// ---------------------------------------------------------------------------
// Pass 2: scores[b,t] = scale*(hidden[b,t].v[b] + c[b]).
// One wave per token; float4 lane-strided dot (134 MB, largely L2-resident).
// grid = B_*T_/8 = 2048 blocks of 256 threads (8 waves).
// ---------------------------------------------------------------------------
__global__ void scores_kernel(const float4* __restrict__ hidden,
                              const float4* __restrict__ vvec,
                              const float* __restrict__ cvec,
                              float* __restrict__ out) {
  const int lane = threadIdx.x & 31;
  const int wave = threadIdx.x >> 5;
  const int g    = blockIdx.x * 8 + wave;  // global token index
  const int b    = g >> 12;                // g / T_
  const float4* hp = hidden + (size_t)g * (D_ / 4);
  const float4* vp = vvec + (size_t)b * (D_ / 4);
  float s = 0.f;
#pragma unroll
  for (int i = 0; i < (D_ / 4) / 32; ++i) {  // 16 iterations
    float4 x = hp[lane + 32 * i];
    float4 y = vp[lane + 32 * i];
    s += x.x * y.x + x.y * y.y + x.z * y.z + x.w * y.w;
  }
#pragma unroll
  for (int off = 16; off > 0; off >>= 1) s += __shfl_xor(s, off, 32);
  if (lane == 0) {
    const float scale = 0.044194173824159216f;  // 512^-0.5
    out[g] = scale * (s + cvec[b]);
  }
}

extern "C" void kernel_launch(void* const* d_in, const int* in_sizes, int n_in,
                              void* d_out, int out_size, void* d_ws, size_t ws_size,
                              hipStream_t stream) {
  (void)in_sizes; (void)n_in; (void)out_size; (void)ws_size;
  const float* hidden = (const float*)d_in[0];
  const float* Wq     = (const float*)d_in[1];
  const float* bq     = (const float*)d_in[2];
  const float* Wk     = (const float*)d_in[3];
  const float* bk     = (const float*)d_in[4];
  float* out = (float*)d_out;

  float* ws      = (float*)d_ws;
  float* partial = ws;                               // B_*NCH_*D_ = 524288 f (2 MB)
  float* hbar    = partial + (size_t)B_ * NCH_ * D_; // B_*D_   = 8192 f
  float* kbar    = hbar + (size_t)B_ * D_;           // B_*DR_  = 2048 f
  float* vvec    = kbar + (size_t)B_ * DR_;          // B_*D_   = 8192 f
  float* cvec    = vvec + (size_t)B_ * D_;           // B_      = 4 f

  colsum_partial<<<B_ * NCH_ * (D_ / 4 / 256), 256, 0, stream>>>(
      (const float4*)hidden, (float4*)partial);
  colsum_final<<<B_ * (D_ / 4 / 256), 256, 0, stream>>>(
      (const float4*)partial, (float4*)hbar);
  kbar_wmma<<<DR_ / 16, 32, 0, stream>>>(hbar, Wk, bk, kbar);
  vproj_wmma<<<D_ / 16 + 1, 32, 0, stream>>>(kbar, Wq, bq, vvec, cvec);
  scores_kernel<<<(B_ * T_) / 8, 256, 0, stream>>>(
      (const float4*)hidden, (const float4*)vvec, cvec, out);
}